// SegCaps_2869038153913
// MI455X (gfx1250) — compile-verified
//
#include <hip/hip_runtime.h>
#include <math.h>

typedef __attribute__((ext_vector_type(2))) float v2f;
typedef __attribute__((ext_vector_type(8))) float v8f;

#define NB 16
#define NR 4096
#define NIC 32
#define NCO 512            // C*OC = 16*32
#define UH_STRIDE 8192     // NB*NCO elements of u_hat per route r

// ---------------------------------------------------------------------------
// Kernel 1: u_hat[r][b][co] = x[b][r][:] @ W[r][co][:]   via V_WMMA_F32_16X16X4_F32
// Also accumulates the iteration-0 sum  s0[b][co] = sum_r u_hat  (uniform softmax;
// the 1/R scale is applied later in squash), saving one 128MB pass over u_hat.
//
// W (256MB) is streamed exactly once -> non-temporal loads so the stream does
// not evict the L2-resident u_hat (128MB of 192MB L2) that the routing passes
// re-read 4x. W[r+1] is prefetched (counter-free global_prefetch_b8) before
// the barrier so HBM latency overlaps the 32 WMMAs of route r.
// ---------------------------------------------------------------------------
__global__ __launch_bounds__(256) void k_uhat_wmma(
    const float* __restrict__ x, const float* __restrict__ W,
    float* __restrict__ uhat, float* __restrict__ s0)
{
  __shared__ float Wlds[32 * 512];   // 64KB, i-major, xor bank-swizzled
  const int tid  = threadIdx.x;
  const int lane = tid & 31;
  const int wave = tid >> 5;         // 8 waves; wave owns 4 N-tiles of 16 cols
  const int half = lane >> 4;        // K-half select per ISA A/B layout
  const int l16  = lane & 15;

  v8f acc[4] = {v8f{}, v8f{}, v8f{}, v8f{}};

  const int rbase = blockIdx.x * 16;
  for (int rr = 0; rr < 16; ++rr) {
    const int r = rbase + rr;

    // Stage W[r] (64KB): coalesced, non-temporal global reads; LDS stored
    // i-major so the WMMA B fragment (K-major) reads are unit-stride. XOR
    // swizzle (co^i) keeps both the scatter-writes and the reads conflict-free.
    const float* wp = W + (size_t)r * (NCO * NIC);
    for (int k = tid; k < NCO * NIC; k += 256) {
      int co = k >> 5, i = k & 31;
      Wlds[i * 512 + (co ^ i)] = __builtin_nontemporal_load(wp + k);
    }

    // Fire-and-forget prefetch of next route's W tile (128 threads x 128B).
    if (rr + 1 < 16 && tid < 128) {
      const float* wn = W + (size_t)(r + 1) * (NCO * NIC);
      __builtin_prefetch(wn + tid * 32, 0, 0);
    }
    __syncthreads();

    // A fragments (x row, 16x32) straight from global (L2-hot, reused 4x here).
    // ISA layout: VGPR0 = K0 (lanes0-15)/K2 (lanes16-31), VGPR1 = K1/K3.
    v2f afrag[8];
#pragma unroll
    for (int kk = 0; kk < 8; ++kk) {
      const int kb = kk * 4 + half * 2;
      const float* xp = x + (size_t)l16 * (NR * NIC) + (size_t)r * NIC + kb;
      afrag[kk].x = xp[0];
      afrag[kk].y = xp[1];
    }

    float* up = uhat + (size_t)r * UH_STRIDE;
#pragma unroll
    for (int t = 0; t < 4; ++t) {
      const int ncol = (wave * 4 + t) * 16 + l16;
      v8f d = {};
#pragma unroll
      for (int kk = 0; kk < 8; ++kk) {
        const int kb = kk * 4 + half * 2;
        v2f bfrag;
        bfrag.x = Wlds[kb * 512 + (ncol ^ kb)];
        bfrag.y = Wlds[(kb + 1) * 512 + (ncol ^ (kb + 1))];
        d = __builtin_amdgcn_wmma_f32_16x16x4_f32(
                false, afrag[kk], false, bfrag, (short)0, d, false, false);
      }
      // D layout: VGPR j holds rows j (lanes 0-15) and j+8 (lanes 16-31)
#pragma unroll
      for (int j = 0; j < 8; ++j)
        up[(j + half * 8) * NCO + ncol] = d[j];
      acc[t] += d;
    }
    __syncthreads();
  }

  // Fold raw s0 partials. Waves own disjoint co ranges -> contention only
  // across the 256 blocks.
#pragma unroll
  for (int t = 0; t < 4; ++t) {
    const int ncol = (wave * 4 + t) * 16 + l16;
#pragma unroll
    for (int j = 0; j < 8; ++j)
      atomicAdd(&s0[(j + half * 8) * NCO + ncol], acc[t][j]);
  }
}

// ---------------------------------------------------------------------------
// squash: v = s^2 * s / ((1+s^2) * sqrt(s^2)), with optional pre-scale
// ---------------------------------------------------------------------------
__global__ __launch_bounds__(256) void k_squash(
    const float* __restrict__ s, float* __restrict__ v, float scale, int n)
{
  int i = blockIdx.x * blockDim.x + threadIdx.x;
  if (i < n) {
    float t  = s[i] * scale;
    float sq = t * t;
    v[i] = sq * t / ((1.0f + sq) * sqrtf(sq));
  }
}

// ---------------------------------------------------------------------------
// a-pass: blog[r][c] += (1/B) * sum_{b,o} u_hat[r][b][c*32+o] * v[b][c*32+o]
// One block per route; thread t = (c,b); float4 streaming reads (L2-resident).
// ---------------------------------------------------------------------------
__global__ __launch_bounds__(256) void k_apass(
    const float* __restrict__ uhat, const float* __restrict__ v,
    float* __restrict__ blog)
{
  __shared__ float red[256];
  const int r = blockIdx.x;
  const int t = threadIdx.x;
  const int c = t >> 4, b = t & 15;

  const float4* u4 = (const float4*)(uhat + (size_t)r * UH_STRIDE + b * NCO + c * 32);
  const float4* v4 = (const float4*)(v + b * NCO + c * 32);
  float sum = 0.0f;
#pragma unroll
  for (int q = 0; q < 8; ++q) {
    float4 a = u4[q], w = v4[q];
    sum += a.x * w.x + a.y * w.y + a.z * w.z + a.w * w.w;
  }
  red[t] = sum;
  __syncthreads();
  if (t < 16) {   // t == c
    float tot = 0.0f;
#pragma unroll
    for (int b2 = 0; b2 < 16; ++b2) tot += red[t * 16 + b2];
    blog[r * 16 + t] += tot * (1.0f / 16.0f);
  }
}

// ---------------------------------------------------------------------------
// softmax over the route axis (4096) per column c; one block per column.
// ---------------------------------------------------------------------------
__global__ __launch_bounds__(256) void k_softmax_col(
    const float* __restrict__ blog, float* __restrict__ crt)
{
  __shared__ float red[256];
  const int c = blockIdx.x;
  const int t = threadIdx.x;

  float m = -INFINITY;
  for (int r = t; r < NR; r += 256) m = fmaxf(m, blog[r * 16 + c]);
  red[t] = m;
  __syncthreads();
  for (int off = 128; off; off >>= 1) {
    if (t < off) red[t] = fmaxf(red[t], red[t + off]);
    __syncthreads();
  }
  m = red[0];
  __syncthreads();

  float ssum = 0.0f;
  for (int r = t; r < NR; r += 256) ssum += expf(blog[r * 16 + c] - m);
  red[t] = ssum;
  __syncthreads();
  for (int off = 128; off; off >>= 1) {
    if (t < off) red[t] += red[t + off];
    __syncthreads();
  }
  const float inv = 1.0f / red[0];
  for (int r = t; r < NR; r += 256)
    crt[r * 16 + c] = expf(blog[r * 16 + c] - m) * inv;
}

// ---------------------------------------------------------------------------
// s-pass: s[b][co] = sum_r crt[r][c(co)] * u_hat[r][b][co]
// 64 blocks x 64 routes; LDS accumulator, one atomic fold at the end.
// ---------------------------------------------------------------------------
__global__ __launch_bounds__(256) void k_spass(
    const float* __restrict__ uhat, const float* __restrict__ crt,
    float* __restrict__ s)
{
  __shared__ float acc[UH_STRIDE];   // 32KB
  __shared__ float crow[16];
  const int t = threadIdx.x;
  for (int k = t; k < UH_STRIDE; k += 256) acc[k] = 0.0f;

  const int rbase = blockIdx.x * 64;
  for (int rr = 0; rr < 64; ++rr) {
    const int r = rbase + rr;
    __syncthreads();
    if (t < 16) crow[t] = crt[r * 16 + t];
    __syncthreads();
    const float* up = uhat + (size_t)r * UH_STRIDE;
    for (int k = t; k < UH_STRIDE; k += 256) {
      const int c = (k & 511) >> 5;
      acc[k] += crow[c] * up[k];   // coalesced u_hat stream (L2-resident)
    }
  }
  __syncthreads();
  for (int k = t; k < UH_STRIDE; k += 256) atomicAdd(&s[k], acc[k]);
}

__global__ __launch_bounds__(256) void k_zero(float* __restrict__ p, int n)
{
  int i = blockIdx.x * blockDim.x + threadIdx.x;
  if (i < n) p[i] = 0.0f;
}

// ---------------------------------------------------------------------------
extern "C" void kernel_launch(void* const* d_in, const int* in_sizes, int n_in,
                              void* d_out, int out_size, void* d_ws, size_t ws_size,
                              hipStream_t stream)
{
  (void)in_sizes; (void)n_in; (void)out_size; (void)ws_size;
  const float* x = (const float*)d_in[0];   // [16,4096,32]
  const float* W = (const float*)d_in[1];   // [4096,16,32,32]
  float* out = (float*)d_out;               // [16,16,32,1] = 8192 f32

  float* uhat = (float*)d_ws;               // [4096][16][512] = 128MB (fits L2)
  float* s    = uhat + (size_t)NR * UH_STRIDE;  // [16][512]
  float* v    = s + UH_STRIDE;                  // [16][512]
  float* blog = v + UH_STRIDE;                  // [4096][16]
  float* crt  = blog + NR * 16;                 // [4096][16]

  // zero routing state (harness does not re-poison between replays)
  k_zero<<<UH_STRIDE / 256, 256, 0, stream>>>(s, UH_STRIDE);
  k_zero<<<NR * 16 / 256, 256, 0, stream>>>(blog, NR * 16);

  // u_hat GEMM (WMMA) + raw iteration-0 sum
  k_uhat_wmma<<<256, 256, 0, stream>>>(x, W, uhat, s);
  k_squash<<<UH_STRIDE / 256, 256, 0, stream>>>(s, v, 1.0f / (float)NR, UH_STRIDE); // v0

  // iteration 0 -> 1
  k_apass<<<NR, 256, 0, stream>>>(uhat, v, blog);
  k_softmax_col<<<16, 256, 0, stream>>>(blog, crt);
  k_zero<<<UH_STRIDE / 256, 256, 0, stream>>>(s, UH_STRIDE);
  k_spass<<<64, 256, 0, stream>>>(uhat, crt, s);
  k_squash<<<UH_STRIDE / 256, 256, 0, stream>>>(s, v, 1.0f, UH_STRIDE);             // v1

  // iteration 1 -> 2
  k_apass<<<NR, 256, 0, stream>>>(uhat, v, blog);
  k_softmax_col<<<16, 256, 0, stream>>>(blog, crt);
  k_zero<<<UH_STRIDE / 256, 256, 0, stream>>>(s, UH_STRIDE);
  k_spass<<<64, 256, 0, stream>>>(uhat, crt, s);
  k_squash<<<UH_STRIDE / 256, 256, 0, stream>>>(s, out, 1.0f, UH_STRIDE);           // v2
}